// BiRNN_2259152798293
// MI455X (gfx1250) — compile-verified
//
#include <hip/hip_runtime.h>
#include <math.h>

// Problem constants (match reference)
#define B_    64
#define T_    512
#define F_    512
#define H_    512
#define KTOT  1024            // F + H (fused [h | x_t] @ [U; W])
#define NWG   8               // workgroups per direction
#define NSL   (H_ / NWG)      // 64 output columns per workgroup
#define KC    32              // K chunk staged in LDS (double buffered)
#define NC    (KTOT / KC)     // 32 chunks per step
#define AP    36              // A LDS pitch (floats): mult of 4 -> 16B rows, bank-skewed
#define THREADS 512           // 16 waves (wave32)

typedef float v2f __attribute__((ext_vector_type(2)));
typedef float v8f __attribute__((ext_vector_type(8)));

__global__ void birnn_init_ws(float* hbuf, unsigned* ctr) {
    int i = blockIdx.x * blockDim.x + threadIdx.x;
    if (i < 2 * 2 * B_ * H_) hbuf[i] = 0.0f;   // both parities, both dirs
    if (i < 2 * T_)          ctr[i]  = 0u;     // per-dir per-step barrier counters
}

// gfx1250 async DMA: global -> LDS, 16B per lane, tracked by ASYNCcnt
__device__ __forceinline__ void async_copy_b128(const float* gsrc, const float* lds_dst) {
    unsigned ldsoff = (unsigned)(uintptr_t)lds_dst;    // flat LDS addr[31:0] = LDS offset
    asm volatile("global_load_async_to_lds_b128 %0, %1, off"
                 :: "v"(ldsoff), "v"(gsrc) : "memory");
}
__device__ __forceinline__ void wait_async0() {
    asm volatile("s_wait_asynccnt 0x0" ::: "memory");
}

__launch_bounds__(THREADS)
__global__ void birnn_scan_kernel(const float* __restrict__ x,
                                  const float* __restrict__ Wf,
                                  const float* __restrict__ Uf,
                                  const float* __restrict__ bf,
                                  const float* __restrict__ Wb,
                                  const float* __restrict__ Ub,
                                  const float* __restrict__ bb,
                                  float* __restrict__ out,
                                  float* __restrict__ hbuf,
                                  unsigned* __restrict__ ctr) {
    const int dir   = blockIdx.x / NWG;        // 0 = forward, 1 = backward
    const int slice = blockIdx.x % NWG;
    const int n0    = slice * NSL;             // global column base of this WG

    const float* W    = dir ? Wb : Wf;
    const float* U    = dir ? Ub : Uf;
    const float* bias = dir ? bb : bf;
    float* h0 = hbuf + (size_t)dir * (2 * B_ * H_);  // parity 0
    float* h1 = h0 + B_ * H_;                        // parity 1
    unsigned* cdir = ctr + dir * T_;

    const int tid  = threadIdx.x;
    const int lane = tid & 31;
    const int wave = tid >> 5;                 // 0..15
    const int half = lane >> 4;                // 0 or 1
    const int lr   = lane & 15;

    // Tile assignment: 64x64 WG tile = 4x4 grid of 16x16 wave tiles
    const int m0   = (wave >> 2) * 16;         // output row base (batch index)
    const int nloc = (wave & 3) * 16;          // column base within WG slice
    const int ngl  = n0 + nloc + lr;           // this lane's output column (0..511)
    const float bv = bias[ngl];                // bias constant across all steps

    // DMA staging assignment (one b128 per thread per matrix per chunk)
    const int arow = tid >> 3;                 // 0..63  (A: batch row)
    const int acol = (tid & 7) * 4;            // 0..28  (A: k offset, 4 floats)
    const int brow = tid >> 4;                 // 0..31  (B: k row)
    const int bcol = (tid & 15) * 4;           // 0..60  (B: n offset, 4 floats)

    __shared__ float Alds[2][B_ * AP];         // 2 x 9216 B
    __shared__ float Blds[2][KC * NSL];        // 2 x 8192 B

    for (int s = 0; s < T_; ++s) {
        const int tx = dir ? (T_ - 1 - s) : s; // x time index read this step
        const float* hprev = (s & 1) ? h1 : h0;
        float*       hnext = (s & 1) ? h0 : h1;

        // Issue async DMA for chunk c into buffer b
        auto issue_chunk = [&](int c, int b) {
            const int kc = c * KC;
            // A chunk: rows = batch, cols = K (h region then x region)
            const float* asrc = (kc < H_)
                ? (hprev + (size_t)arow * H_ + kc + acol)
                : (x + (size_t)arow * (T_ * F_) + (size_t)tx * F_ + (kc - H_) + acol);
            async_copy_b128(asrc, &Alds[b][arow * AP + acol]);
            // B chunk: rows = K, cols = this WG's 64-column slice of [U; W]
            const int gk = kc + brow;
            const float* bsrc = (gk < H_)
                ? (U + (size_t)gk * H_ + n0 + bcol)
                : (W + (size_t)(gk - H_) * H_ + n0 + bcol);
            async_copy_b128(bsrc, &Blds[b][brow * NSL + bcol]);
        };

        v8f acc = {0.f, 0.f, 0.f, 0.f, 0.f, 0.f, 0.f, 0.f};

        issue_chunk(0, 0);                     // prologue: stage chunk 0

        for (int c = 0; c < NC; ++c) {
            const int b = c & 1;
            wait_async0();                     // my chunk-c DMA landed in LDS
            __syncthreads();                   // all waves landed; buf b^1 reads done
            if (c + 1 < NC) issue_chunk(c + 1, b ^ 1);   // overlaps with WMMAs below

            const float* Ab = &Alds[b][0];
            const float* Bb = &Blds[b][0];
            // 16x16x4 fp32 WMMA over the chunk.
            // A frag (16x4): lane l, vgpr v -> A[M = l&15][K = kq + v]
            // B frag (4x16): lane l, vgpr v -> B[K = kq + v][N = l&15]
            // with kq = kk + 2*(l>>4)
            #pragma unroll
            for (int kk = 0; kk < KC; kk += 4) {
                const int kq = kk + 2 * half;
                v2f av, bvv;
                av.x  = Ab[(m0 + lr) * AP + kq];
                av.y  = Ab[(m0 + lr) * AP + kq + 1];
                bvv.x = Bb[kq * NSL + nloc + lr];
                bvv.y = Bb[(kq + 1) * NSL + nloc + lr];
                acc = __builtin_amdgcn_wmma_f32_16x16x4_f32(
                    /*neg_a=*/false, av, /*neg_b=*/false, bvv,
                    /*c_mod=*/(short)0, acc, /*reuse_a=*/false, /*reuse_b=*/false);
            }
        }

        // --- epilogue: bias + tanh, write h_next and output slab
        // C/D layout: vgpr r, lane l -> row m0 + r + 8*(l>>4), col nloc + (l&15)
        #pragma unroll
        for (int r = 0; r < 8; ++r) {
            const int m = m0 + r + 8 * half;
            const float v = tanhf(acc[r] + bv);
            hnext[m * H_ + ngl] = v;
            out[(size_t)m * (T_ * 2 * H_) + (size_t)s * (2 * H_) + dir * H_ + ngl] = v;
        }

        // --- cross-WG barrier for this direction (release/acquire on L2)
        __syncthreads();
        if (tid == 0) {
            __threadfence();
            __hip_atomic_fetch_add(&cdir[s], 1u, __ATOMIC_RELEASE, __HIP_MEMORY_SCOPE_AGENT);
            while (__hip_atomic_load(&cdir[s], __ATOMIC_ACQUIRE, __HIP_MEMORY_SCOPE_AGENT)
                   < (unsigned)NWG) { }
        }
        __syncthreads();
    }
}

extern "C" void kernel_launch(void* const* d_in, const int* in_sizes, int n_in,
                              void* d_out, int out_size, void* d_ws, size_t ws_size,
                              hipStream_t stream) {
    const float* x  = (const float*)d_in[0];
    const float* Wf = (const float*)d_in[1];
    const float* Uf = (const float*)d_in[2];
    const float* bf = (const float*)d_in[3];
    const float* Wb = (const float*)d_in[4];
    const float* Ub = (const float*)d_in[5];
    const float* bb = (const float*)d_in[6];
    float* out = (float*)d_out;

    float*    hbuf = (float*)d_ws;                                   // 512 KB (2 dirs x 2 parities x 64x512)
    unsigned* ctr  = (unsigned*)((char*)d_ws + (size_t)2 * 2 * B_ * H_ * sizeof(float)); // 4 KB

    const int initN = 2 * 2 * B_ * H_;
    birnn_init_ws<<<(initN + 255) / 256, 256, 0, stream>>>(hbuf, ctr);

    birnn_scan_kernel<<<2 * NWG, THREADS, 0, stream>>>(
        x, Wf, Uf, bf, Wb, Ub, bb, out, hbuf, ctr);
}